// KG_GAT_75969381531866
// MI455X (gfx1250) — compile-verified
//
#include <hip/hip_runtime.h>

#define NNODES 50000
#define NEDGES 200000
#define IN_DIM 768
#define HID_DIM 256
#define OUT_DIM 64
#define H1 4
#define NEG_SLOPE 0.2f
#define LN_EPS 1e-5f

typedef __attribute__((ext_vector_type(2))) float v2f;
typedef __attribute__((ext_vector_type(8))) float v8f;

// ---------- order-preserving float<->uint encoding (for deterministic segment max) ----------
__device__ __forceinline__ unsigned f32_enc(float f) {
  unsigned u = __float_as_uint(f);
  return (u & 0x80000000u) ? ~u : (u | 0x80000000u);
}
__device__ __forceinline__ float f32_dec(unsigned e) {
  unsigned b = (e & 0x80000000u) ? (e & 0x7FFFFFFFu) : ~e;
  return __uint_as_float(b);
}

// ---------- CDNA5 async global->LDS copies (ASYNCcnt-tracked, ISA ch.10/15.18) ----------
// Flat pointers to LDS carry the wave-relative byte offset in their low 32 bits,
// which is exactly what the VDST operand of global_load_async_to_lds wants.
__device__ __forceinline__ unsigned lds_off(const void* p) {
  return (unsigned)(unsigned long long)p;
}
__device__ __forceinline__ void async_copy_b128(unsigned lds_addr, const float* gaddr) {
  asm volatile("global_load_async_to_lds_b128 %0, %1, off" ::"v"(lds_addr), "v"(gaddr)
               : "memory");
}
__device__ __forceinline__ void async_copy_b64(unsigned lds_addr, const float* gaddr) {
  asm volatile("global_load_async_to_lds_b64 %0, %1, off" ::"v"(lds_addr), "v"(gaddr)
               : "memory");
}
template <int N>
__device__ __forceinline__ void wait_asynccnt() {
  asm volatile("s_wait_asynccnt %0" ::"n"(N) : "memory");
}

// ---------- scatter-set: last edge id per source node wins ----------
__global__ void k_last_edge(const int* __restrict__ src, int* __restrict__ last) {
  int e = blockIdx.x * blockDim.x + threadIdx.x;
  if (e < NEDGES) atomicMax(&last[src[e]], e);
}

__global__ void k_xmod(const float* __restrict__ x, const int* __restrict__ last,
                       const int* __restrict__ etype, const float* __restrict__ eemb,
                       float* __restrict__ xmod) {
  size_t idx = (size_t)blockIdx.x * blockDim.x + threadIdx.x;
  if (idx >= (size_t)NNODES * IN_DIM) return;
  int n = (int)(idx / IN_DIM);
  int c = (int)(idx - (size_t)n * IN_DIM);
  float v = x[idx];
  int le = last[n];
  if (le >= 0) v += eemb[(size_t)etype[le] * IN_DIM + c];
  xmod[idx] = v;
}

// ---------- f32 WMMA GEMM with double-buffered async-LDS staging ----------
// Block: 16 output rows x NCOLS. One wave per 16x16 N-tile. Per K-chunk (KC=64):
//   A slab 16xKC and B slab KCxNCOLS are staged global->LDS with
//   GLOBAL_LOAD_ASYNC_TO_LDS (once per block, not once per wave), double-buffered
//   on ASYNCcnt, then all waves feed v_wmma_f32_16x16x4_f32 from LDS.
// LDS padding: A row stride KC+4 (bank rotation 4), B row stride NCOLS+16
// (rotation 16) -> the two half-wave row-pairs of each fragment land on disjoint banks.
template <int K, int NCOLS, int KC>
__global__ __launch_bounds__(32 * (NCOLS / 16)) void k_gemm_wmma(
    const float* __restrict__ A, const float* __restrict__ B, float* __restrict__ C) {
  constexpr int THREADS = 32 * (NCOLS / 16);
  constexpr int ASTR = KC + 4;       // floats
  constexpr int BSTR = NCOLS + 16;   // floats
  constexpr int BL = (KC * NCOLS / 4) / THREADS;  // b128 loads per thread per chunk
  constexpr int AL = (8 * KC) / THREADS;          // b64 loads per thread per chunk
  constexpr int L = BL + AL;                      // async instrs per wave per chunk
  constexpr int NCHUNKS = K / KC;

  __shared__ float Ab[2][16 * ASTR];
  __shared__ float Bb[2][KC * BSTR];

  const int t = threadIdx.x;
  const int wave = t >> 5;
  const int lane = t & 31;
  const int half = lane >> 4;
  const int l16 = lane & 15;
  const int mbase = blockIdx.x * 16;
  const int nbase = wave * 16;
  const float* Ag = A + (size_t)mbase * K;

  // issue all async copies for one K-chunk into buffer `buf`
  auto issue = [&](int chunk, int buf) {
    const int kc = chunk * KC;
    unsigned boff = lds_off(&Bb[buf][0]);
    unsigned aoff = lds_off(&Ab[buf][0]);
#pragma unroll
    for (int j = 0; j < BL; ++j) {
      int s = j * THREADS + t;
      int k = s / (NCOLS / 4);
      int c4 = s - k * (NCOLS / 4);
      async_copy_b128(boff + (unsigned)((k * BSTR + c4 * 4) * 4),
                      B + (size_t)(kc + k) * NCOLS + c4 * 4);
    }
#pragma unroll
    for (int j = 0; j < AL; ++j) {
      int s = j * THREADS + t;
      int m = s / (KC / 2);
      int c2 = s - m * (KC / 2);
      async_copy_b64(aoff + (unsigned)((m * ASTR + c2 * 2) * 4),
                     Ag + (size_t)m * K + kc + c2 * 2);
    }
  };

  v8f acc = {0.f, 0.f, 0.f, 0.f, 0.f, 0.f, 0.f, 0.f};

  issue(0, 0);
  for (int i = 0; i < NCHUNKS; ++i) {
    if (i + 1 < NCHUNKS) {
      issue(i + 1, (i + 1) & 1);
      wait_asynccnt<L>();  // chunk i fully in LDS; chunk i+1 still streaming
    } else {
      wait_asynccnt<0>();
    }
    __syncthreads();

    const float* Ap = &Ab[i & 1][0];
    const float* Bp = &Bb[i & 1][0];
#pragma unroll
    for (int k = 0; k < KC; k += 4) {
      v2f a = *(const v2f*)&Ap[l16 * ASTR + k + half * 2];  // ds_load_b64
      v2f b;
      b.x = Bp[(k + half * 2) * BSTR + nbase + l16];
      b.y = Bp[(k + half * 2 + 1) * BSTR + nbase + l16];
      acc = __builtin_amdgcn_wmma_f32_16x16x4_f32(false, a, false, b, (short)0, acc,
                                                  false, false);
    }
    __syncthreads();
  }

  float* crow = C + (size_t)(mbase + half * 8) * NCOLS + nbase + l16;
#pragma unroll
  for (int r = 0; r < 8; ++r) crow[(size_t)r * NCOLS] = acc[r];
}

// ---------- per-(node,head) attention logits: dot over 64 dims, one wave each ----------
template <int HEADS>
__global__ void k_att_logits(const float* __restrict__ hp, const float* __restrict__ a_src,
                             const float* __restrict__ a_dst, float* __restrict__ al_s,
                             float* __restrict__ al_d) {
  int w = (blockIdx.x * blockDim.x + threadIdx.x) >> 5;
  int lane = threadIdx.x & 31;
  if (w >= NNODES * HEADS) return;  // whole wave exits together
  int n = w / HEADS, h = w - n * HEADS;
  const float* hv = hp + (size_t)n * (HEADS * 64) + h * 64 + lane * 2;
  const float* as = a_src + h * 64 + lane * 2;
  const float* ad = a_dst + h * 64 + lane * 2;
  float ss = hv[0] * as[0] + hv[1] * as[1];
  float sd = hv[0] * ad[0] + hv[1] * ad[1];
#pragma unroll
  for (int off = 16; off > 0; off >>= 1) {
    ss += __shfl_xor(ss, off, 32);
    sd += __shfl_xor(sd, off, 32);
  }
  if (lane == 0) {
    al_s[w] = ss;
    al_d[w] = sd;
  }
}

// ---------- edge pass 1: leaky-relu score + deterministic segment max ----------
template <int HEADS>
__global__ void k_edge_score_max(const int* __restrict__ src, const int* __restrict__ dst,
                                 const float* __restrict__ al_s, const float* __restrict__ al_d,
                                 float* __restrict__ ex, unsigned* __restrict__ emax) {
  int idx = blockIdx.x * blockDim.x + threadIdx.x;
  if (idx >= NEDGES * HEADS) return;
  int e = idx / HEADS, h = idx - e * HEADS;
  int s = src[e], d = dst[e];
  float v = al_s[s * HEADS + h] + al_d[d * HEADS + h];
  v = v > 0.0f ? v : NEG_SLOPE * v;
  ex[idx] = v;  // raw score, overwritten by exp in pass 2
  atomicMax(&emax[d * HEADS + h], f32_enc(v));
}

// ---------- edge pass 2: exp(s - max) and denominator ----------
template <int HEADS>
__global__ void k_edge_exp_denom(const int* __restrict__ dst, float* __restrict__ ex,
                                 const unsigned* __restrict__ emax, float* __restrict__ denom) {
  int idx = blockIdx.x * blockDim.x + threadIdx.x;
  if (idx >= NEDGES * HEADS) return;
  int e = idx / HEADS, h = idx - e * HEADS;
  int d = dst[e];
  float v = expf(ex[idx] - f32_dec(emax[d * HEADS + h]));
  ex[idx] = v;
  atomicAdd(&denom[d * HEADS + h], v);
}

// ---------- edge pass 3: weighted scatter-aggregate (one edge per block) ----------
template <int CH, int HEADS>
__global__ void k_aggregate(const int* __restrict__ src, const int* __restrict__ dst,
                            const float* __restrict__ hp, const float* __restrict__ ex,
                            const float* __restrict__ denom, float* __restrict__ out) {
  int e = blockIdx.x;
  int c = threadIdx.x;  // blockDim.x == CH
  int h = c >> 6;       // head size is always 64
  int s = src[e], d = dst[e];
  float alpha = ex[e * HEADS + h] / denom[d * HEADS + h];
  atomicAdd(&out[(size_t)d * CH + c], hp[(size_t)s * CH + c] * alpha);
}

// ---------- bias + LayerNorm (+ optional ELU), one node per block ----------
template <int CH, bool DO_ELU>
__global__ void k_ln(const float* __restrict__ in, const float* __restrict__ bias,
                     const float* __restrict__ g, const float* __restrict__ be,
                     float* __restrict__ out) {
  __shared__ float red[CH];
  int n = blockIdx.x, c = threadIdx.x;
  float v = in[(size_t)n * CH + c] + bias[c];
  red[c] = v;
  __syncthreads();
#pragma unroll
  for (int s = CH / 2; s > 0; s >>= 1) {
    if (c < s) red[c] += red[c + s];
    __syncthreads();
  }
  float mu = red[0] * (1.0f / CH);
  __syncthreads();
  float dv = v - mu;
  red[c] = dv * dv;
  __syncthreads();
#pragma unroll
  for (int s = CH / 2; s > 0; s >>= 1) {
    if (c < s) red[c] += red[c + s];
    __syncthreads();
  }
  float var = red[0] * (1.0f / CH);
  float y = dv * rsqrtf(var + LN_EPS) * g[c] + be[c];
  if (DO_ELU) y = y > 0.0f ? y : expm1f(y);
  out[(size_t)n * CH + c] = y;
}

extern "C" void kernel_launch(void* const* d_in, const int* in_sizes, int n_in,
                              void* d_out, int out_size, void* d_ws, size_t ws_size,
                              hipStream_t stream) {
  const float* x = (const float*)d_in[0];
  const int* eidx = (const int*)d_in[1];
  const int* etype = (const int*)d_in[2];
  const float* eemb = (const float*)d_in[3];
  const float* W1 = (const float*)d_in[4];
  const float* a_src1 = (const float*)d_in[5];
  const float* a_dst1 = (const float*)d_in[6];
  const float* b1 = (const float*)d_in[7];
  const float* g1 = (const float*)d_in[8];
  const float* be1 = (const float*)d_in[9];
  const float* W2 = (const float*)d_in[10];
  const float* a_src2 = (const float*)d_in[11];
  const float* a_dst2 = (const float*)d_in[12];
  const float* b2 = (const float*)d_in[13];
  const float* g2 = (const float*)d_in[14];
  const float* be2 = (const float*)d_in[15];
  const int* src = eidx;           // edge_index[0]
  const int* dst = eidx + NEDGES;  // edge_index[1]

  // ---- carve workspace (~290 MB) ----
  char* p = (char*)d_ws;
  auto carve = [&](size_t bytes) -> void* {
    void* r = (void*)p;
    p += (bytes + 255) & ~(size_t)255;
    return r;
  };
  float* xmod = (float*)carve((size_t)NNODES * IN_DIM * 4);
  float* h1p = (float*)carve((size_t)NNODES * HID_DIM * 4);   // proj, then reused for LN output
  float* out1 = (float*)carve((size_t)NNODES * HID_DIM * 4);  // aggregated layer-1
  float* h2p = (float*)carve((size_t)NNODES * OUT_DIM * 4);
  float* out2 = (float*)carve((size_t)NNODES * OUT_DIM * 4);
  float* al_s1 = (float*)carve((size_t)NNODES * H1 * 4);
  float* al_d1 = (float*)carve((size_t)NNODES * H1 * 4);
  unsigned* emax1 = (unsigned*)carve((size_t)NNODES * H1 * 4);
  float* denom1 = (float*)carve((size_t)NNODES * H1 * 4);
  float* ex1 = (float*)carve((size_t)NEDGES * H1 * 4);
  float* al_s2 = (float*)carve((size_t)NNODES * 4);
  float* al_d2 = (float*)carve((size_t)NNODES * 4);
  unsigned* emax2 = (unsigned*)carve((size_t)NNODES * 4);
  float* denom2 = (float*)carve((size_t)NNODES * 4);
  float* ex2 = (float*)carve((size_t)NEDGES * 4);
  int* last = (int*)carve((size_t)NNODES * 4);

  // ---- per-call init (graph-capture safe) ----
  hipMemsetAsync(last, 0xFF, (size_t)NNODES * 4, stream);  // -1
  hipMemsetAsync(emax1, 0, (size_t)NNODES * H1 * 4, stream);
  hipMemsetAsync(denom1, 0, (size_t)NNODES * H1 * 4, stream);
  hipMemsetAsync(out1, 0, (size_t)NNODES * HID_DIM * 4, stream);
  hipMemsetAsync(emax2, 0, (size_t)NNODES * 4, stream);
  hipMemsetAsync(denom2, 0, (size_t)NNODES * 4, stream);
  hipMemsetAsync(out2, 0, (size_t)NNODES * OUT_DIM * 4, stream);

  // ---- x_mod = x ; x_mod[src] = x[src] + edge_emb[edge_type]  (last edge wins) ----
  k_last_edge<<<(NEDGES + 255) / 256, 256, 0, stream>>>(src, last);
  k_xmod<<<(int)(((size_t)NNODES * IN_DIM + 255) / 256), 256, 0, stream>>>(x, last, etype, eemb,
                                                                           xmod);

  // ---- layer 1: async-staged WMMA GEMM, attention, aggregate, LN+ELU ----
  k_gemm_wmma<IN_DIM, HID_DIM, 64>
      <<<NNODES / 16, 32 * (HID_DIM / 16), 0, stream>>>(xmod, W1, h1p);
  k_att_logits<H1><<<(NNODES * H1 + 7) / 8, 256, 0, stream>>>(h1p, a_src1, a_dst1, al_s1, al_d1);
  k_edge_score_max<H1><<<(NEDGES * H1 + 255) / 256, 256, 0, stream>>>(src, dst, al_s1, al_d1,
                                                                      ex1, emax1);
  k_edge_exp_denom<H1><<<(NEDGES * H1 + 255) / 256, 256, 0, stream>>>(dst, ex1, emax1, denom1);
  k_aggregate<HID_DIM, H1><<<NEDGES, HID_DIM, 0, stream>>>(src, dst, h1p, ex1, denom1, out1);
  k_ln<HID_DIM, true><<<NNODES, HID_DIM, 0, stream>>>(out1, b1, g1, be1, h1p);  // h1p := h1f

  // ---- layer 2: async-staged WMMA GEMM, attention (1 head), aggregate, LN ----
  k_gemm_wmma<HID_DIM, OUT_DIM, 64>
      <<<NNODES / 16, 32 * (OUT_DIM / 16), 0, stream>>>(h1p, W2, h2p);
  k_att_logits<1><<<(NNODES + 7) / 8, 256, 0, stream>>>(h2p, a_src2, a_dst2, al_s2, al_d2);
  k_edge_score_max<1><<<(NEDGES + 255) / 256, 256, 0, stream>>>(src, dst, al_s2, al_d2, ex2,
                                                                emax2);
  k_edge_exp_denom<1><<<(NEDGES + 255) / 256, 256, 0, stream>>>(dst, ex2, emax2, denom2);
  k_aggregate<OUT_DIM, 1><<<NEDGES, OUT_DIM, 0, stream>>>(src, dst, h2p, ex2, denom2, out2);
  k_ln<OUT_DIM, false><<<NNODES, OUT_DIM, 0, stream>>>(out2, b2, g2, be2, (float*)d_out);
}